// unitary_simple_80650895884948
// MI455X (gfx1250) — compile-verified
//
#include <hip/hip_runtime.h>

// ---------- types ----------
typedef __attribute__((ext_vector_type(2))) float v2f;
typedef __attribute__((ext_vector_type(4))) float v4f;
typedef __attribute__((ext_vector_type(8))) float v8f;

#define SEGMENTS 64
#define SU2NDIM  16
#define BATCH    128
#define WAVES_PER_BLOCK 8
#define MAT_ELEMS 256            // 16x16 floats

// ---------------------------------------------------------------------------
// Layouts (wave32):
//  A-operand (16x4 chunk): element (m,k): lane = m + 16*(kk>=2), vgpr = kk%2
//  B-operand (4x16 chunk): element (k,n): lane = n + 16*(kk>=2), vgpr = kk%2
//  C/D accumulator:        element (m,n): lane = n + 16*(m>=8),  vgpr = m%8
// ---------------------------------------------------------------------------

// half-wave swap: lane <-> lane^16 (DS_SWIZZLE_B32, no LDS memory access)
__device__ inline float swap16(float v) {
    union { float f; int i; } u; u.f = v;
    u.i = __builtin_amdgcn_ds_swizzle(u.i, 0x401F);  // xor=0x10, and=0x1f
    return u.f;
}

struct Bop { v2f b[4]; };

// Convert a C/D-layout register matrix into B-operand chunk registers using
// only half-wave swizzles + selects (no LDS memory traffic).
//  B(k=0): rows 0/2, 1/3   B(k=1): rows 4/6, 5/7
//  B(k=2): rows 8/10, 9/11 B(k=3): rows 12/14, 13/15
__device__ inline Bop cd_to_bop(const v8f& cd, int h) {
    const float s0 = swap16(cd[0]), s1 = swap16(cd[1]);
    const float s2 = swap16(cd[2]), s3 = swap16(cd[3]);
    const float s4 = swap16(cd[4]), s5 = swap16(cd[5]);
    const float s6 = swap16(cd[6]), s7 = swap16(cd[7]);
    Bop o;
    o.b[0].x = h ? s2    : cd[0];  o.b[0].y = h ? s3    : cd[1];
    o.b[1].x = h ? s6    : cd[4];  o.b[1].y = h ? s7    : cd[5];
    o.b[2].x = h ? cd[2] : s0;     o.b[2].y = h ? cd[3] : s1;
    o.b[3].x = h ? cd[6] : s4;     o.b[3].y = h ? cd[7] : s5;
    return o;
}

// Load the A-operand (4 chunk register-pairs) from a row-major LDS matrix.
__device__ inline void load_aop(const float* __restrict__ M, int h, int mn,
                                v2f a[4]) {
#pragma unroll
    for (int k = 0; k < 4; ++k) {
        const int ka = 4 * k + 2 * h;
        a[k].x = M[mn * 16 + ka];
        a[k].y = M[mn * 16 + ka + 1];
    }
}

// 16x16x16 f32 matmul: 4 chained V_WMMA_F32_16X16X4_F32, result in C/D regs.
__device__ inline v8f wave_mm(const v2f a[4], const Bop& B) {
    v8f c = {};
#pragma unroll
    for (int k = 0; k < 4; ++k)
        c = __builtin_amdgcn_wmma_f32_16x16x4_f32(
                false, a[k], false, B.b[k], (short)0, c, false, false);
    return c;
}

__device__ inline void store_cd(float* __restrict__ M, const v8f& c,
                                int h, int mn) {
#pragma unroll
    for (int r = 0; r < 8; ++r)
        M[(r + 8 * h) * 16 + mn] = c[r];
}

// ---------------------------------------------------------------------------
// One wave handles one (b, s) pair:
//   U = prod_{j=15..0} expm(-x[127-b, 16*s + j])
// expm: scaling s=2 (M = A/4, spectral radius ~0.1, 1-norm <= ~0.4),
// degree-8 Taylor via Horner (7 matmuls), then 2 squarings.
// Remainder <= 0.4^9/9! ~ 7e-10  (below f32 eps).
// ---------------------------------------------------------------------------
__global__ __launch_bounds__(256)
void unitary_chain_wmma_kernel(const float* __restrict__ x,
                               float* __restrict__ out) {
    __shared__ __align__(16) float lA[WAVES_PER_BLOCK][MAT_ELEMS];
    __shared__ __align__(16) float lT[WAVES_PER_BLOCK][MAT_ELEMS];

    const int lane = (int)(threadIdx.x & 31);
    const int w    = (int)(threadIdx.x >> 5);
    const int h    = lane >> 4;     // half-wave id
    const int mn   = lane & 15;

    const int pair = (int)blockIdx.x * WAVES_PER_BLOCK + w;   // 0..8191
    const int b    = pair >> 6;
    const int s    = pair & 63;
    const int bb   = (BATCH - 1) - b;                         // reversed batch

    float* A = lA[w];
    float* T = lT[w];

    const float inv_scale = 0.25f;          // 2^-2 scaling, 2 squarings
    v8f U;                                  // running product, C/D registers

    for (int j = 0; j < 16; ++j) {
        // ---- stage A = -x[bb, 16*s + j] into LDS (coalesced b128) ----
        const size_t moff = (((size_t)bb * (SEGMENTS * SU2NDIM)
                              + (size_t)(SU2NDIM * s + j)) << 8);
        const v4f* __restrict__ src4 = (const v4f*)(x + moff);
        v4f* __restrict__ a4 = (v4f*)A;
        a4[2 * lane]     = -src4[2 * lane];
        a4[2 * lane + 1] = -src4[2 * lane + 1];

        // ---- P0 = I + (A/4)/8  (C/D registers) ----
        v8f P;
#pragma unroll
        for (int r = 0; r < 8; ++r) {
            const float diag = ((r + 8 * h) == mn) ? 1.0f : 0.0f;
            P[r] = A[(r + 8 * h) * 16 + mn] * (inv_scale / 8.0f) + diag;
        }

        // A-operand registers (constant across the Horner loop)
        v2f aA[4];
        load_aop(A, h, mn, aA);

        // ---- Horner: P = I + (A/(4k)) @ P, k = 7..1  (no LDS traffic) ----
#pragma unroll
        for (int k = 7; k >= 1; --k) {
            const Bop Bp = cd_to_bop(P, h);
            const v8f t  = wave_mm(aA, Bp);
            const float coef = inv_scale / (float)k;
#pragma unroll
            for (int r = 0; r < 8; ++r) {
                const float diag = ((r + 8 * h) == mn) ? 1.0f : 0.0f;
                P[r] = t[r] * coef + diag;
            }
        }

        // ---- 2 squarings: P = P @ P  (A-op via LDS, B-op via swizzle) ----
#pragma unroll
        for (int sq = 0; sq < 2; ++sq) {
            store_cd(T, P, h, mn);
            v2f aP[4];
            load_aop(T, h, mn, aP);
            const Bop Bp = cd_to_bop(P, h);
            P = wave_mm(aP, Bp);
        }

        // ---- chain: U = E @ U  (U never leaves registers) ----
        if (j == 0) {
            U = P;                           // E0 @ I
        } else {
            store_cd(T, P, h, mn);
            v2f aP[4];
            load_aop(T, h, mn, aP);
            const Bop Bu = cd_to_bop(U, h);
            U = wave_mm(aP, Bu);
        }
    }

    // ---- write result: bounce through LDS for coalesced float4 stores ----
    store_cd(A, U, h, mn);
    v4f* __restrict__ dst4 = (v4f*)(out + ((size_t)pair << 8));
    const v4f* __restrict__ u4 = (const v4f*)A;
    dst4[2 * lane]     = u4[2 * lane];
    dst4[2 * lane + 1] = u4[2 * lane + 1];
}

// ---------------------------------------------------------------------------
extern "C" void kernel_launch(void* const* d_in, const int* in_sizes, int n_in,
                              void* d_out, int out_size, void* d_ws, size_t ws_size,
                              hipStream_t stream) {
    (void)in_sizes; (void)n_in; (void)out_size; (void)d_ws; (void)ws_size;
    const float* x = (const float*)d_in[0];
    float* out = (float*)d_out;

    const int pairs  = BATCH * SEGMENTS;                  // 8192 waves
    const int blocks = pairs / WAVES_PER_BLOCK;           // 1024 blocks
    hipLaunchKernelGGL(unitary_chain_wmma_kernel,
                       dim3(blocks), dim3(WAVES_PER_BLOCK * 32), 0, stream,
                       x, out);
}